// NTM_30983894073573
// MI455X (gfx1250) — compile-verified
//
#include <hip/hip_runtime.h>
#include <math.h>

typedef __attribute__((ext_vector_type(2))) float v2f;
typedef __attribute__((ext_vector_type(8))) float v8f;

namespace {
constexpr int T_  = 16;
constexpr int B_  = 32;
constexpr int NI_ = 64;
constexpr int NO_ = 64;
constexpr int H_  = 256;
constexpr int N_  = 2048;
constexpr int M_  = 64;
constexpr int S_  = 3;
constexpr int P_  = 3*M_ + S_ + 3;   // 198
constexpr int G_  = 4*H_;            // 1024 gate width
constexpr int KF_ = H_ + M_;         // 320  [h, r]
constexpr int NBLK_ = 16;            // n-chunks per batch in read/update pass
constexpr int PT_ = (P_ + 15)/16;    // 13 n-tiles for the p GEMM
}

__device__ __forceinline__ float sigmoidf_(float x){ return 1.f/(1.f + __expf(-x)); }
__device__ __forceinline__ float softplusf_(float x){ return (x > 20.f) ? x : log1pf(__expf(x)); }

// ---------------------------------------------------------------------------
// init: w = w0 / sum(w0) per batch. one block per batch.
__global__ __launch_bounds__(256) void w_init_k(const float* __restrict__ w0,
                                                float* __restrict__ w)
{
    __shared__ float red[256];
    const int b = blockIdx.x, tid = threadIdx.x;
    float s = 0.f;
    for (int n = tid; n < N_; n += 256) s += w0[b*N_ + n];
    red[tid] = s; __syncthreads();
    for (int k = 128; k > 0; k >>= 1) { if (tid < k) red[tid] += red[tid + k]; __syncthreads(); }
    const float inv = 1.f / red[0];
    for (int n = tid; n < N_; n += 256) w[b*N_ + n] = w0[b*N_ + n] * inv;
}

// ---------------------------------------------------------------------------
// gates = [r|x|h] @ [Wih | Whh]^T + bih + bhh   (32 x 1024, K = 384)
// 8 waves/block, one 16x16 tile per wave, f32 WMMA 16x16x4.
// K-loop split into three branch-free segments so the compiler can unroll
// and clause-pack the WMMA pipeline (cat is never materialized).
__global__ __launch_bounds__(256) void gates_wmma_k(const float* __restrict__ r,
                                                    const float* __restrict__ xt,
                                                    const float* __restrict__ h,
                                                    const float* __restrict__ Wih,
                                                    const float* __restrict__ Whh,
                                                    const float* __restrict__ bih,
                                                    const float* __restrict__ bhh,
                                                    float* __restrict__ gates)
{
    const int lane = threadIdx.x & 31;
    const int tile = blockIdx.x*8 + (threadIdx.x >> 5);   // 0..127
    const int tn = tile & 63;            // output-col tile (1024/16)
    const int tm = tile >> 6;            // batch tile (32/16)
    const int col = tn*16 + (lane & 15);
    const int row = tm*16 + (lane & 15);
    const int khi = (lane >> 4) << 1;    // 0 (lanes 0-15) / 2 (lanes 16-31)

    const float* ar = r  + row*M_;       // k in [0,64)
    const float* ax = xt + row*NI_;      // k in [64,128)
    const float* ah = h  + row*H_;       // k in [128,384)
    const float* wi = Wih + col*(NI_ + M_);
    const float* wh = Whh + col*H_;

    v8f acc = {};
    // segment 1: r x Wih[:, 0:64)
    for (int k = 0; k < M_; k += 4) {
        const int k0 = k + khi;
        v2f a, bb;
        a.x  = ar[k0];     a.y  = ar[k0 + 1];
        bb.x = wi[k0];     bb.y = wi[k0 + 1];
        acc = __builtin_amdgcn_wmma_f32_16x16x4_f32(false, a, false, bb,
                                                    (short)0, acc, false, false);
    }
    // segment 2: x x Wih[:, 64:128)
    for (int k = 0; k < NI_; k += 4) {
        const int k0 = k + khi;
        v2f a, bb;
        a.x  = ax[k0];          a.y  = ax[k0 + 1];
        bb.x = wi[M_ + k0];     bb.y = wi[M_ + k0 + 1];
        acc = __builtin_amdgcn_wmma_f32_16x16x4_f32(false, a, false, bb,
                                                    (short)0, acc, false, false);
    }
    // segment 3: h x Whh
    for (int k = 0; k < H_; k += 4) {
        const int k0 = k + khi;
        v2f a, bb;
        a.x  = ah[k0];     a.y  = ah[k0 + 1];
        bb.x = wh[k0];     bb.y = wh[k0 + 1];
        __builtin_prefetch(wh + k0 + 32, 0, 1);
        acc = __builtin_amdgcn_wmma_f32_16x16x4_f32(false, a, false, bb,
                                                    (short)0, acc, false, false);
    }

    const float bias = bih[col] + bhh[col];
    const int mhi = (lane >> 4) << 3;
#pragma unroll
    for (int v = 0; v < 8; ++v) {
        const int m = tm*16 + v + mhi;
        gates[m*G_ + col] = acc[v] + bias;
    }
}

// ---------------------------------------------------------------------------
// LSTM pointwise (split order i, f, g, o)
__global__ __launch_bounds__(256) void lstm_pw_k(const float* __restrict__ gates,
                                                 float* __restrict__ h,
                                                 float* __restrict__ c)
{
    const int idx = blockIdx.x*256 + threadIdx.x;   // 0 .. B*H-1
    if (idx >= B_*H_) return;
    const int b = idx / H_, j = idx % H_;
    const float ig = gates[b*G_ + 0*H_ + j];
    const float fg = gates[b*G_ + 1*H_ + j];
    const float gg = gates[b*G_ + 2*H_ + j];
    const float og = gates[b*G_ + 3*H_ + j];
    const float cn = sigmoidf_(fg)*c[idx] + sigmoidf_(ig)*tanhf(gg);
    c[idx] = cn;
    h[idx] = sigmoidf_(og)*tanhf(cn);
}

// ---------------------------------------------------------------------------
// p = h @ Wp + bp   (32 x 198, K = 256). Wp row-major (256, 198). 8 waves/block.
__global__ __launch_bounds__(256) void p_wmma_k(const float* __restrict__ h,
                                                const float* __restrict__ Wp,
                                                const float* __restrict__ bp,
                                                float* __restrict__ p)
{
    const int lane = threadIdx.x & 31;
    const int tile = blockIdx.x*8 + (threadIdx.x >> 5);
    if (tile >= 2*PT_) return;           // wave-uniform guard (EXEC all-ones inside)
    const int tn = tile % PT_;
    const int tm = tile / PT_;
    const int col  = tn*16 + (lane & 15);
    const int lcol = (col < P_) ? col : (P_ - 1);     // clamp loads
    const int row  = tm*16 + (lane & 15);
    const int khi  = (lane >> 4) << 1;

    v8f acc = {};
    for (int k = 0; k < H_; k += 4) {
        const int k0 = k + khi;
        v2f a, bb;
        a.x = h[row*H_ + k0];
        a.y = h[row*H_ + k0 + 1];
        bb.x = Wp[(k0    )*P_ + lcol];
        bb.y = Wp[(k0 + 1)*P_ + lcol];
        acc = __builtin_amdgcn_wmma_f32_16x16x4_f32(false, a, false, bb,
                                                    (short)0, acc, false, false);
    }
    if (col < P_) {
        const float bias = bp[col];
        const int mhi = (lane >> 4) << 3;
#pragma unroll
        for (int v = 0; v < 8; ++v) {
            const int m = tm*16 + v + mhi;
            p[m*P_ + col] = acc[v] + bias;
        }
    }
}

// ---------------------------------------------------------------------------
// Fused addressing (one block per batch):
//  activate head params from raw p -> kappa(+eps) in LDS -> knorm ->
//  cosine similarity over this batch's 2048 memory rows (L2-resident) ->
//  softmax -> interpolate with old w -> circular shift -> sharpen -> normalize.
// raw p layout: [0:64)=kappa [64]=beta [65]=g [66:69)=s [69]=gamma [70:134)=e [134:198)=a
__global__ __launch_bounds__(256) void addressing_k(const float* __restrict__ mem,
                                                    const float* __restrict__ p,
                                                    float* __restrict__ w)
{
    __shared__ float red[256];
    __shared__ float wg[N_];
    __shared__ float kap_s[M_];
    const int b = blockIdx.x, tid = threadIdx.x;
    const float* pb = p + b*P_;

    const float beta = softplusf_(pb[M_]);
    const float g    = sigmoidf_(pb[M_ + 1]);
    const float s0r = pb[M_+2], s1r = pb[M_+3], s2r = pb[M_+4];
    const float smx = fmaxf(s0r, fmaxf(s1r, s2r));
    const float e0 = __expf(s0r - smx), e1 = __expf(s1r - smx), e2 = __expf(s2r - smx);
    const float sinv = 1.f / (e0 + e1 + e2);
    const float s0 = e0*sinv, s1 = e1*sinv, s2 = e2*sinv;
    const float gamma = 1.f + softplusf_(pb[M_ + 2 + S_]);

    if (tid < M_) kap_s[tid] = pb[tid] + 1e-12f;
    __syncthreads();

    float kn2 = 0.f;
#pragma unroll
    for (int m = 0; m < M_; ++m) kn2 += kap_s[m]*kap_s[m];
    const float knorm = sqrtf(kn2);

    // similarity (beta * cosine) + running max, into LDS
    float mx = -3.4e38f;
    for (int n = tid; n < N_; n += 256) {
        const float* mrow = mem + ((size_t)b*N_ + n)*M_;
        float num = 0.f, nn = 0.f;
#pragma unroll
        for (int m = 0; m < M_; m += 4) {
            const float4 mv = *(const float4*)(mrow + m);
            const float m0 = mv.x + 1e-12f, m1 = mv.y + 1e-12f;
            const float m2 = mv.z + 1e-12f, m3 = mv.w + 1e-12f;
            num += m0*kap_s[m] + m1*kap_s[m+1] + m2*kap_s[m+2] + m3*kap_s[m+3];
            nn  += m0*m0 + m1*m1 + m2*m2 + m3*m3;
        }
        const float sv = beta * num / fmaxf(sqrtf(nn)*knorm, 1e-8f);
        wg[n] = sv;
        mx = fmaxf(mx, sv);
    }
    red[tid] = mx; __syncthreads();
    for (int k = 128; k > 0; k >>= 1) { if (tid < k) red[tid] = fmaxf(red[tid], red[tid+k]); __syncthreads(); }
    const float mval = red[0]; __syncthreads();

    // exp + sum
    float sum = 0.f;
    for (int n = tid; n < N_; n += 256) { const float e = __expf(wg[n] - mval); wg[n] = e; sum += e; }
    red[tid] = sum; __syncthreads();
    for (int k = 128; k > 0; k >>= 1) { if (tid < k) red[tid] += red[tid+k]; __syncthreads(); }
    const float inv = 1.f / red[0]; __syncthreads();

    // interpolate with old w
    for (int n = tid; n < N_; n += 256)
        wg[n] = g*(wg[n]*inv) + (1.f - g)*w[b*N_ + n];
    __syncthreads();

    // circular shift + sharpen
    float psum = 0.f;
    for (int n = tid; n < N_; n += 256) {
        const float wh = s0*wg[(n + N_ - 1) & (N_-1)] + s1*wg[n] + s2*wg[(n + 1) & (N_-1)];
        const float wp = powf(wh, gamma);
        w[b*N_ + n] = wp;
        psum += wp;
    }
    red[tid] = psum; __syncthreads();
    for (int k = 128; k > 0; k >>= 1) { if (tid < k) red[tid] += red[tid+k]; __syncthreads(); }
    const float inv2 = 1.f / (red[0] + 1e-12f);
    for (int n = tid; n < N_; n += 256) w[b*N_ + n] *= inv2;
}

// ---------------------------------------------------------------------------
// fused read (with OLD mem) + erase/add write; e/a activated inline from raw p.
// One block = (batch, 128-row n chunk). Deterministic partials into rpart.
__global__ __launch_bounds__(256) void read_update_k(const float* __restrict__ w,
                                                     const float* __restrict__ p,
                                                     float* __restrict__ mem,
                                                     float* __restrict__ rpart)
{
    __shared__ float red[256];
    const int blk = blockIdx.x;
    const int b  = blk / NBLK_;
    const int nb = blk % NBLK_;
    const int tid = threadIdx.x;
    const int m  = tid & (M_ - 1);
    const int nl = tid >> 6;                               // 0..3
    const float ev = sigmoidf_(p[b*P_ + M_ + 3 + S_ + m]); // erase
    const float av = tanhf   (p[b*P_ + 2*M_ + 3 + S_ + m]);// add
    float acc = 0.f;
    const int nbase = nb * (N_ / NBLK_);
    for (int i = 0; i < (N_ / NBLK_) / 4; ++i) {
        const int n = nbase + i*4 + nl;
        const float wn = w[b*N_ + n];
        const size_t off = ((size_t)b*N_ + n)*M_ + m;
        const float mv = mem[off];
        acc += wn * mv;                                    // read uses old mem
        mem[off] = mv * (1.f - wn*ev) + wn*av;             // erase/add
    }
    red[tid] = acc; __syncthreads();
    if (nl == 0)
        rpart[((size_t)b*NBLK_ + nb)*M_ + m] = red[m] + red[m+64] + red[m+128] + red[m+192];
}

// ---------------------------------------------------------------------------
// fused: r = reduce(rpart) (to LDS + global for next step), then
// out_t = sigmoid([h, r] @ Wfc^T + bfc) via 8 WMMA waves in one block.
// K-loop split at the h|r seam: global-A segment then LDS-A segment.
__global__ __launch_bounds__(256) void fc_fused_k(const float* __restrict__ h,
                                                  const float* __restrict__ rpart,
                                                  const float* __restrict__ Wfc,
                                                  const float* __restrict__ bfc,
                                                  float* __restrict__ r_out,
                                                  float* __restrict__ out_t)
{
    __shared__ float rsh[B_*M_];                 // 2048 floats
    const int tid = threadIdx.x;
    for (int i = tid; i < B_*M_; i += 256) {
        const int b = i / M_, m = i % M_;
        float s = 0.f;
#pragma unroll
        for (int j = 0; j < NBLK_; ++j) s += rpart[((size_t)b*NBLK_ + j)*M_ + m];
        rsh[i] = s;
        r_out[i] = s;
    }
    __syncthreads();

    const int lane = tid & 31;
    const int wave = tid >> 5;                   // 0..7  (2 m-tiles x 4 n-tiles)
    const int tn = wave & 3;
    const int tm = wave >> 2;
    const int col = tn*16 + (lane & 15);
    const int row = tm*16 + (lane & 15);
    const int khi = (lane >> 4) << 1;

    const float* ah = h + row*H_;
    const float* wf = Wfc + col*KF_;
    const float* rl = rsh + row*M_;

    v8f acc = {};
    // segment 1: h part (global A)
    for (int k = 0; k < H_; k += 4) {
        const int k0 = k + khi;
        v2f a, bb;
        a.x  = ah[k0];     a.y  = ah[k0 + 1];
        bb.x = wf[k0];     bb.y = wf[k0 + 1];
        __builtin_prefetch(wf + k0 + 32, 0, 1);
        acc = __builtin_amdgcn_wmma_f32_16x16x4_f32(false, a, false, bb,
                                                    (short)0, acc, false, false);
    }
    // segment 2: r part (LDS A)
    for (int k = 0; k < M_; k += 4) {
        const int k0 = k + khi;
        v2f a, bb;
        a.x  = rl[k0];          a.y  = rl[k0 + 1];
        bb.x = wf[H_ + k0];     bb.y = wf[H_ + k0 + 1];
        acc = __builtin_amdgcn_wmma_f32_16x16x4_f32(false, a, false, bb,
                                                    (short)0, acc, false, false);
    }

    const float bias = bfc[col];
    const int mhi = (lane >> 4) << 3;
#pragma unroll
    for (int v = 0; v < 8; ++v) {
        const int m = tm*16 + v + mhi;
        out_t[m*NO_ + col] = sigmoidf_(acc[v] + bias);
    }
}

// ---------------------------------------------------------------------------
extern "C" void kernel_launch(void* const* d_in, const int* in_sizes, int n_in,
                              void* d_out, int out_size, void* d_ws, size_t ws_size,
                              hipStream_t stream)
{
    const float* x    = (const float*)d_in[0];   // (T,B,NI)
    const float* mem0 = (const float*)d_in[1];   // (B,N,M)
    const float* w0   = (const float*)d_in[2];   // (B,N)
    const float* r0   = (const float*)d_in[3];   // (B,M)
    const float* h0   = (const float*)d_in[4];   // (B,H)
    const float* c0   = (const float*)d_in[5];   // (B,H)
    const float* Wih  = (const float*)d_in[6];   // (4H, NI+M)
    const float* Whh  = (const float*)d_in[7];   // (4H, H)
    const float* bih  = (const float*)d_in[8];
    const float* bhh  = (const float*)d_in[9];
    const float* Wp   = (const float*)d_in[10];  // (H, P) row-major
    const float* bp   = (const float*)d_in[11];
    const float* Wfc  = (const float*)d_in[12];  // (NO, H+M)
    const float* bfc  = (const float*)d_in[13];

    float* outs = (float*)d_out;                 // (T,B,NO)
    float* memW = outs + T_*B_*NO_;              // (B,N,M): working memory lives in d_out

    float* ws    = (float*)d_ws;
    float* gates = ws;                 // B*4H     = 32768
    float* h     = gates + B_*G_;      // 8192
    float* c     = h     + B_*H_;      // 8192
    float* r     = c     + B_*H_;      // 2048
    float* w     = r     + B_*M_;      // 65536
    float* p     = w     + B_*N_;      // 6336
    float* rpart = p     + B_*P_;      // B*NBLK*M = 32768

    // re-initialize full state every call (deterministic; inputs untouched)
    hipMemcpyAsync(memW, mem0, sizeof(float)*(size_t)B_*N_*M_, hipMemcpyDeviceToDevice, stream);
    hipMemcpyAsync(h, h0, sizeof(float)*B_*H_, hipMemcpyDeviceToDevice, stream);
    hipMemcpyAsync(c, c0, sizeof(float)*B_*H_, hipMemcpyDeviceToDevice, stream);
    hipMemcpyAsync(r, r0, sizeof(float)*B_*M_, hipMemcpyDeviceToDevice, stream);
    w_init_k<<<B_, 256, 0, stream>>>(w0, w);

    for (int t = 0; t < T_; ++t) {
        gates_wmma_k <<<16, 256, 0, stream>>>(r, x + (size_t)t*B_*NI_, h,
                                              Wih, Whh, bih, bhh, gates);
        lstm_pw_k    <<<(B_*H_ + 255)/256, 256, 0, stream>>>(gates, h, c);
        p_wmma_k     <<<(2*PT_ + 7)/8, 256, 0, stream>>>(h, Wp, bp, p);
        addressing_k <<<B_, 256, 0, stream>>>(memW, p, w);
        read_update_k<<<B_*NBLK_, 256, 0, stream>>>(w, p, memW, rpart);
        fc_fused_k   <<<1, 256, 0, stream>>>(h, rpart, Wfc, bfc, r,
                                             outs + (size_t)t*B_*NO_);
    }
}